// H_second_conv_20040317403399
// MI455X (gfx1250) — compile-verified
//
#include <hip/hip_runtime.h>

typedef __attribute__((ext_vector_type(2))) float v2f;
typedef __attribute__((ext_vector_type(4))) float v4f;
typedef __attribute__((ext_vector_type(8))) float v8f;

#define B_   64
#define C_   64
#define H_   32
#define P_   496
#define HK_  560
#define KB_  140          // HK_/4 K-steps of 4
#define XST  34           // padded x row stride (dwords)

__global__ __launch_bounds__(256) void
H_second_conv_wmma_kernel(const float* __restrict__ x,
                          const float* __restrict__ w,
                          const int*   __restrict__ idx_a,
                          const int*   __restrict__ idx_b,
                          float* __restrict__ out)
{
    __shared__ float        x_lds[B_ * XST];     //  8,704 B
    __shared__ unsigned int pair_pk[P_];         //  1,984 B (pairs packed as u16x2)
    __shared__ float        wfrag[KB_ * 32 * 4]; // 71,680 B (B frags, nt-interleaved)
    __shared__ float        red[4 * 2 * 256];    //  8,192 B (split-K partials)

    const int group = blockIdx.x;   // 0..2047
    const int c = group >> 5;
    const int i = group & (H_ - 1);
    const int t = threadIdx.x;      // 0..255

    // ---- async stage x rows (64 x 32 f32) into padded LDS rows ----
    for (int v = t; v < B_ * (H_ / 4); v += 256) {
        const int b  = v >> 3;
        const int jq = v & 7;
        const float* src = x + (((size_t)b * C_ + c) * H_ + i) * H_ + jq * 4;
        const unsigned ldsoff = (unsigned)(uintptr_t)&x_lds[b * XST + jq * 4];
        asm volatile("global_load_async_to_lds_b128 %0, %1, off"
                     :: "v"(ldsoff), "v"(src) : "memory");
    }

    // ---- pack cross-index pairs: one u64 = (ia0,ib0,ia1,ib1) as u16 ----
    for (int p2 = t; p2 < P_ / 2; p2 += 256) {
        const int p = 2 * p2;
        pair_pk[p]     = ((unsigned)idx_a[p]     & 0xffffu) | ((unsigned)idx_b[p]     << 16);
        pair_pk[p + 1] = ((unsigned)idx_a[p + 1] & 0xffffu) | ((unsigned)idx_b[p + 1] << 16);
    }

    // ---- stage weight[c][i] (560x32 f32) swizzled to nt-interleaved B-frag layout ----
    // lane L, dword d: d = nt*2+vv holds B[K = 2*(L>=16)+vv][N = nt*16 + L%16]
    {
        const float* wg = w + ((size_t)(c * H_ + i)) * HK_ * H_;
        for (int v = t; v < HK_ * (H_ / 4); v += 256) {  // 4480 float4 loads
            const int k  = v >> 3;
            const int nq = v & 7;
            const float4 val = *(const float4*)(wg + k * H_ + nq * 4);
            const int kb = k >> 2, r = k & 3;
            const int hi = r >> 1, vv = r & 1;
            const float vals[4] = {val.x, val.y, val.z, val.w};
            #pragma unroll
            for (int q = 0; q < 4; ++q) {
                const int n  = nq * 4 + q;
                const int nt = n >> 4, Lm = n & 15;
                const int L  = hi * 16 + Lm;
                wfrag[kb * 128 + L * 4 + nt * 2 + vv] = vals[q];
            }
        }
    }
    asm volatile("s_wait_asynccnt 0x0" ::: "memory");
    __syncthreads();

    // ---- GEMM: 8 waves = 4 M-tiles x 2 K-halves; each wave does both N-tiles ----
    const int wave  = t >> 5;
    const int lane  = t & 31;
    const int mt    = wave & 3;        // M tile (batch/16)
    const int kh    = wave >> 2;       // K half
    const int brow  = mt * 16 + (lane & 15);
    const int khalf = (lane < 16) ? 0 : 2;   // A layout half
    const float* xrow = &x_lds[brow * XST];

    v8f acc0 = {0.f, 0.f, 0.f, 0.f, 0.f, 0.f, 0.f, 0.f};
    v8f acc1 = {0.f, 0.f, 0.f, 0.f, 0.f, 0.f, 0.f, 0.f};

#define BFRAGS(kb)                                                     \
    const v4f bv = *(const v4f*)&wfrag[(kb) * 128 + lane * 4];         \
    v2f b0; b0.x = bv.x; b0.y = bv.y;                                  \
    v2f b1; b1.x = bv.z; b1.y = bv.w;

#define DO_WMMA()                                                      \
    acc0 = __builtin_amdgcn_wmma_f32_16x16x4_f32(false, a, false, b0,  \
                                (short)0, acc0, false, false);         \
    acc1 = __builtin_amdgcn_wmma_f32_16x16x4_f32(false, a, false, b1,  \
                                (short)0, acc1, false, false);

#define CROSS_BODY(kb)                                                 \
    {                                                                  \
        const int p2 = ((kb) * 4 - 64 + khalf) >> 1;                   \
        const uint2 pk = *(const uint2*)&pair_pk[2 * p2];              \
        v2f a;                                                         \
        a.x = xrow[pk.x & 0xffffu] * xrow[pk.x >> 16];                 \
        a.y = xrow[pk.y & 0xffffu] * xrow[pk.y >> 16];                 \
        BFRAGS(kb)                                                     \
        DO_WMMA()                                                      \
    }

    if (kh == 0) {
        // phase 1: k = 0..31 -> feat = x
        #pragma unroll
        for (int kb = 0; kb < 8; ++kb) {
            const int k0 = kb * 4 + khalf;
            v2f a; a.x = xrow[k0]; a.y = xrow[k0 + 1];
            BFRAGS(kb)
            DO_WMMA()
        }
        // phase 2: k = 32..63 -> feat = x*x
        #pragma unroll
        for (int kb = 8; kb < 16; ++kb) {
            const int k0 = (kb - 8) * 4 + khalf;
            const float u = xrow[k0], s = xrow[k0 + 1];
            v2f a; a.x = u * u; a.y = s * s;
            BFRAGS(kb)
            DO_WMMA()
        }
        // phase 3a: cross terms, first K half
        for (int kb = 16; kb < 70; ++kb) CROSS_BODY(kb)
    } else {
        // phase 3b: cross terms, second K half
        for (int kb = 70; kb < KB_; ++kb) CROSS_BODY(kb)
    }

    // ---- split-K reduction via LDS, then store ----
    if (kh == 1) {
        float* r0 = &red[(mt * 2 + 0) * 256 + lane * 8];
        float* r1 = &red[(mt * 2 + 1) * 256 + lane * 8];
        v4f t0; t0.x = acc0[0]; t0.y = acc0[1]; t0.z = acc0[2]; t0.w = acc0[3];
        v4f t1; t1.x = acc0[4]; t1.y = acc0[5]; t1.z = acc0[6]; t1.w = acc0[7];
        v4f t2; t2.x = acc1[0]; t2.y = acc1[1]; t2.z = acc1[2]; t2.w = acc1[3];
        v4f t3; t3.x = acc1[4]; t3.y = acc1[5]; t3.z = acc1[6]; t3.w = acc1[7];
        ((v4f*)r0)[0] = t0; ((v4f*)r0)[1] = t1;
        ((v4f*)r1)[0] = t2; ((v4f*)r1)[1] = t3;
    }
    __syncthreads();
    if (kh == 0) {
        const v4f* r0 = (const v4f*)&red[(mt * 2 + 0) * 256 + lane * 8];
        const v4f* r1 = (const v4f*)&red[(mt * 2 + 1) * 256 + lane * 8];
        const v4f p00 = r0[0], p01 = r0[1], p10 = r1[0], p11 = r1[1];
        #pragma unroll
        for (int q = 0; q < 4; ++q) {
            acc0[q]     += p00[q];
            acc0[q + 4] += p01[q];
            acc1[q]     += p10[q];
            acc1[q + 4] += p11[q];
        }
        // C/D layout: vgpr r, lane L -> M = r + 8*(L>=16), N = L%16
        const int Mbase = (lane < 16) ? 0 : 8;
        const int N     = lane & 15;
        #pragma unroll
        for (int r = 0; r < 8; ++r) {
            const int b = mt * 16 + Mbase + r;
            float* orow = out + (((size_t)b * C_ + c) * H_ + i) * H_;
            orow[N]      = acc0[r];
            orow[16 + N] = acc1[r];
        }
    }
#undef CROSS_BODY
#undef DO_WMMA
#undef BFRAGS
}

extern "C" void kernel_launch(void* const* d_in, const int* in_sizes, int n_in,
                              void* d_out, int out_size, void* d_ws, size_t ws_size,
                              hipStream_t stream) {
    (void)in_sizes; (void)n_in; (void)d_ws; (void)ws_size; (void)out_size;
    const float* x      = (const float*)d_in[0];
    const float* weight = (const float*)d_in[1];
    const int*   idx_a  = (const int*)d_in[2];
    const int*   idx_b  = (const int*)d_in[3];
    float* out = (float*)d_out;

    dim3 grid(C_ * H_);   // one workgroup per (channel, row) group
    dim3 block(256);      // 8 wave32
    H_second_conv_wmma_kernel<<<grid, block, 0, stream>>>(x, weight, idx_a, idx_b, out);
}